// SemblanceLayer_62105227100227
// MI455X (gfx1250) — compile-verified
//
#include <hip/hip_runtime.h>

// Semblance (box-window normalized energy) over (2,256,256,384) fp32.
// Bandwidth-bound: ~402MB HBM traffic -> ~17us floor @ 23.3 TB/s.
// CDNA5 path: async global->LDS copies (ASYNCcnt), s_wait_asynccnt, NT stores.

#define FLT_MAX_F 3.402823466e+38f

#if defined(__has_builtin)
#  if __has_builtin(__builtin_amdgcn_s_wait_asynccnt)
#    define WAIT_ASYNC_BUILTIN 1
#  endif
#endif

typedef float v4f __attribute__((ext_vector_type(4)));

// ISA 15.18.3: GLOBAL_LOAD_ASYNC_TO_LDS_B128 — VDST holds the LDS byte
// address (wave-relative), VADDR the 64-bit global address, SADDR = off.
// The clang builtin uses LangAS::cuda_device/cuda_shared pointer params that
// HIP source can't spell, so use inline asm (backend support is proven by the
// builtin's existence).
__device__ __forceinline__ void async_copy_b128(const float* g, const float* l) {
#if defined(__HIP_DEVICE_COMPILE__)
  unsigned lds = (unsigned)(unsigned long long)(const void*)l;  // addr[31:0] == LDS addr
  asm volatile("global_load_async_to_lds_b128 %0, %1, off"
               :: "v"(lds), "v"(g) : "memory");
#endif
}

__device__ __forceinline__ void wait_async_lds() {
#if defined(__HIP_DEVICE_COMPILE__)
#if defined(WAIT_ASYNC_BUILTIN)
  __builtin_amdgcn_s_wait_asynccnt(0);
#else
  asm volatile("s_wait_asynccnt 0" ::: "memory");
#endif
#endif
}

// Block: 256 threads (8 waves, wave32). Tile: (b, i, 8 j-rows, all 384 depths).
__global__ __launch_bounds__(256, 2)
void semblance_kernel(const float* __restrict__ x, float* __restrict__ out) {
  __shared__ float  xs[12 * 384];   // j-halo tile: rows J0-2..J0+9 (clamped), 18KB
  __shared__ float2 sq[8 * 384];    // per (j,t): {s^2, sum x^2 over j-window}, 24KB

  const int tid = threadIdx.x;
  const int J0  = blockIdx.x * 8;   // j-tile base (32 tiles)
  const int i   = blockIdx.y;       // axis-1 index (w0==1 -> passthrough)
  const int b   = blockIdx.z;
  const size_t rowbase = ((size_t)(b * 256 + i)) * 256;  // row index of (b,i,j=0)

  // ---- Stage A: async copy 12 rows x 384 floats into LDS (1152 x 16B) ----
  #pragma unroll
  for (int it = 0; it < 5; ++it) {
    int idx = it * 256 + tid;
    if (idx < 1152) {
      int row   = idx / 96;                 // 96 16B-chunks per row
      int chunk = idx - row * 96;
      int g = J0 - 2 + row;                 // clamp; OOB rows are zeroed in stage B
      g = g < 0 ? 0 : (g > 255 ? 255 : g);
      const float* gp = x + (rowbase + (size_t)g) * 384 + chunk * 4;
      async_copy_b128(gp, &xs[row * 384 + chunk * 4]);
    }
  }
  wait_async_lds();
  __syncthreads();

  const int jl   = tid >> 5;        // wave index == local j row (0..7)
  const int lane = tid & 31;
  const int jg   = J0 + jl;

  // ---- Stage B: 5-tap crossline window -> pack {s^2, q} into LDS ----
  #pragma unroll
  for (int k = 0; k < 12; ++k) {
    int t = lane + (k << 5);
    float s = 0.f, q = 0.f;
    #pragma unroll
    for (int dj = 0; dj < 5; ++dj) {
      int jj = jg + dj - 2;                       // zero-padding at j edges
      bool ok = (jj >= 0) & (jj < 256);
      float v = ok ? xs[(jl + dj) * 384 + t] : 0.f;
      s += v;
      q = fmaf(v, v, q);
    }
    sq[jl * 384 + t] = make_float2(s * s, q);
  }
  // sq row jl is produced and consumed by the SAME wave (tid>>5); the backend's
  // s_wait_dscnt covers the LDS RAW, so no block-wide barrier is needed here.

  // ---- Stage C: 20-tap depth window (sliding), normalize, nan_to_num ----
  const int t0 = lane * 12;                       // 32 lanes x 12 depths = 384
  float2 w[32];
  #pragma unroll
  for (int m = 0; m < 32; ++m) {                  // taps t0-10 .. t0+21
    int tin = t0 - 10 + m;
    bool ok = (tin >= 0) & (tin < 384);
    w[m] = ok ? sq[jl * 384 + tin] : make_float2(0.f, 0.f);
  }
  float num = 0.f, den = 0.f;
  #pragma unroll
  for (int m = 0; m < 20; ++m) { num += w[m].x; den += w[m].y; }

  const int cl = (jg < 2 ? jg : 2) + ((255 - jg) < 2 ? (255 - jg) : 2) + 1;
  const float c1 = (float)cl;                     // valid-count normalizer

  float res[12];
  #pragma unroll
  for (int e = 0; e < 12; ++e) {
    float dn = den * c1;
    float r  = num / dn;
    r = (r != r) ? 1.0f : r;                      // nan -> fill (1.0)
    r = (r >  FLT_MAX_F) ?  FLT_MAX_F : r;        // +inf -> max finite
    r = (r < -FLT_MAX_F) ? -FLT_MAX_F : r;        // -inf -> lowest finite
    res[e] = r;
    num += w[e + 20].x - w[e].x;                  // slide window by one depth
    den += w[e + 20].y - w[e].y;
  }

  // 3 x b128 nontemporal stores (output is 201MB streamed once; > 192MB L2)
  float* op = out + (rowbase + (size_t)jg) * 384 + t0;
  #pragma unroll
  for (int vq = 0; vq < 3; ++vq) {
    v4f val = { res[4 * vq + 0], res[4 * vq + 1], res[4 * vq + 2], res[4 * vq + 3] };
    __builtin_nontemporal_store(val, (v4f*)(op + 4 * vq));
  }
}

extern "C" void kernel_launch(void* const* d_in, const int* in_sizes, int n_in,
                              void* d_out, int out_size, void* d_ws, size_t ws_size,
                              hipStream_t stream) {
  (void)in_sizes; (void)n_in; (void)out_size; (void)d_ws; (void)ws_size;
  const float* x = (const float*)d_in[0];
  float* out = (float*)d_out;
  dim3 grid(256 / 8, 256, 2);   // (j-tiles, i, b)
  dim3 block(256);
  semblance_kernel<<<grid, block, 0, stream>>>(x, out);
}